// HeteroGnn_14353780703956
// MI455X (gfx1250) — compile-verified
//
#include <hip/hip_runtime.h>
#include <hip/hip_bf16.h>
#include <math.h>
#include <stdint.h>

#define DCH 128

typedef float v2f __attribute__((ext_vector_type(2)));
typedef float v8f __attribute__((ext_vector_type(8)));

__device__ __forceinline__ float gelu_tanh(float x) {
    // jax.nn.gelu default (approximate=True)
    float x3 = x * x * x;
    float t = tanhf(0.7978845608028654f * (x + 0.044715f * x3));
    return 0.5f * x * (1.0f + t);
}

// ---------------------------------------------------------------------------
// C[numRows x 128] = A[numRows x 128] @ W[128 x 128]  (+bias, +gelu optional)
// One wave per 16-row M-tile. W staged into LDS via CDNA5 async global->LDS
// (GLOBAL_LOAD_ASYNC_TO_LDS_B128, ASYNCcnt). Core: V_WMMA_F32_16X16X4_F32.
// ---------------------------------------------------------------------------
__global__ __launch_bounds__(256) void gemm_f32_wmma(
    const float* __restrict__ A, const float* __restrict__ W,
    float* __restrict__ C, const float* __restrict__ bias,
    int fuseGelu, int numRows)
{
    __shared__ float sW[DCH * DCH];   // 64 KB of the 320 KB WGP LDS

    const int tid = threadIdx.x;
    // Async staging: each of the 256 lanes pulls 16 x 16B = 256 B, 64 KB total.
    // Memory -> LDS directly, no VGPR round-trip; tracked by ASYNCcnt.
    {
        const unsigned ldsBase = (unsigned)(uintptr_t)(&sW[0]); // LDS offset in bits [31:0]
        #pragma unroll
        for (int i = 0; i < 16; ++i) {
            const int idx4 = i * 256 + tid;                 // float4 index
            const unsigned lds_off = ldsBase + (unsigned)idx4 * 16u;
            const float* gp = W + (size_t)idx4 * 4;
            asm volatile("global_load_async_to_lds_b128 %0, %1, off"
                         :: "v"(lds_off), "v"(gp)
                         : "memory");
        }
        asm volatile("s_wait_asynccnt 0x0" ::: "memory");
    }
    __syncthreads();

    const int wave  = tid >> 5;
    const int lane  = tid & 31;
    const int mTile = blockIdx.x * 8 + wave;
    if (mTile * 16 >= numRows) return;

    const int half = lane >> 4;          // 0: lanes 0-15, 1: lanes 16-31
    const int l16  = lane & 15;
    int rowA = mTile * 16 + l16;         // A-frag row for this lane (ISA layout)
    if (rowA >= numRows) rowA = numRows - 1;   // clamp (tail tiles)

    // Preload the 16x128 A strip into registers: 32 k-steps of v2f per lane.
    v2f aF[32];
    const float* Arow = A + (size_t)rowA * DCH;
    #pragma unroll
    for (int kk = 0; kk < 32; ++kk) {
        const int k = kk * 4 + 2 * half;  // VGPR0: K=k(+2*half), VGPR1: K+1
        aF[kk].x = Arow[k];
        aF[kk].y = Arow[k + 1];
    }

    #pragma unroll 1
    for (int nt = 0; nt < 8; ++nt) {
        const int col = nt * 16 + l16;    // C/B column = lane%16
        v8f acc = {0.f, 0.f, 0.f, 0.f, 0.f, 0.f, 0.f, 0.f};
        #pragma unroll
        for (int kk = 0; kk < 32; ++kk) {
            const int k = kk * 4 + 2 * half;
            v2f bF;
            bF.x = sW[(k)     * DCH + col];
            bF.y = sW[(k + 1) * DCH + col];
            acc = __builtin_amdgcn_wmma_f32_16x16x4_f32(
                /*neg_a=*/false, aF[kk], /*neg_b=*/false, bF,
                /*c_mod=*/(short)0, acc, /*reuse_a=*/false, /*reuse_b=*/false);
        }
        const float bb = bias ? bias[col] : 0.0f;
        #pragma unroll
        for (int v = 0; v < 8; ++v) {
            const int row = mTile * 16 + v + 8 * half;   // C layout: M = v (+8)
            if (row < numRows) {
                float o = acc[v] + bb;
                if (fuseGelu) o = gelu_tanh(o);
                C[(size_t)row * DCH + col] = o;
            }
        }
    }
}

// ---------------------------------------------------------------------------
// scatter-sum: acc[dst[e], :] += hw[src[e], :]  — one wave / edge, f32 atomics
// ---------------------------------------------------------------------------
__global__ __launch_bounds__(256) void edge_scatter(
    const float* __restrict__ hw, const int* __restrict__ src,
    const int* __restrict__ dst, float* __restrict__ acc, int E)
{
    const long long gt = (long long)blockIdx.x * blockDim.x + threadIdx.x;
    const int e    = (int)(gt >> 5);
    const int lane = (int)(gt & 31);
    if (e >= E) return;
    const int s = src[e];
    const int d = dst[e];
    const float4 v = ((const float4*)(hw + (size_t)s * DCH))[lane];
    float* p = acc + (size_t)d * DCH + lane * 4;
    atomicAdd(p + 0, v.x);
    atomicAdd(p + 1, v.y);
    atomicAdd(p + 2, v.z);
    atomicAdd(p + 3, v.w);
}

// ---------------------------------------------------------------------------
// per-node: out = gelu(layer_norm(acc + b0 + b1) * gamma + beta), wave32/node
// ---------------------------------------------------------------------------
__global__ __launch_bounds__(256) void ln_gelu_kernel(
    const float* __restrict__ acc, const float* __restrict__ bias0,
    const float* __restrict__ bias1, const float* __restrict__ gamma,
    const float* __restrict__ beta, float* __restrict__ out, int Nn)
{
    const long long gt = (long long)blockIdx.x * blockDim.x + threadIdx.x;
    const int node = (int)(gt >> 5);
    const int lane = (int)(gt & 31);
    if (node >= Nn) return;
    const int c = lane * 4;

    float4 v = *(const float4*)(acc + (size_t)node * DCH + c);
    v.x += bias0[c + 0] + bias1[c + 0];
    v.y += bias0[c + 1] + bias1[c + 1];
    v.z += bias0[c + 2] + bias1[c + 2];
    v.w += bias0[c + 3] + bias1[c + 3];

    float s = v.x + v.y + v.z + v.w;
    #pragma unroll
    for (int off = 16; off > 0; off >>= 1) s += __shfl_xor(s, off, 32);
    const float mean = s * (1.0f / DCH);

    const float dx = v.x - mean, dy = v.y - mean, dz = v.z - mean, dw = v.w - mean;
    float sq = dx * dx + dy * dy + dz * dz + dw * dw;
    #pragma unroll
    for (int off = 16; off > 0; off >>= 1) sq += __shfl_xor(sq, off, 32);
    const float rstd = rsqrtf(sq * (1.0f / DCH) + 1e-5f);

    float4 o;
    o.x = gelu_tanh(dx * rstd * gamma[c + 0] + beta[c + 0]);
    o.y = gelu_tanh(dy * rstd * gamma[c + 1] + beta[c + 1]);
    o.z = gelu_tanh(dz * rstd * gamma[c + 2] + beta[c + 2]);
    o.w = gelu_tanh(dw * rstd * gamma[c + 3] + beta[c + 3]);
    *(float4*)(out + (size_t)node * DCH + c) = o;
}

// ---------------------------------------------------------------------------
// head tail: out[n, 0:2] = g[n, :] @ W1[128x2] + b1 — wave32 per node
// ---------------------------------------------------------------------------
__global__ __launch_bounds__(256) void head_out_kernel(
    const float* __restrict__ g, const float* __restrict__ W1,
    const float* __restrict__ b1, float* __restrict__ out, int Nn)
{
    const long long gt = (long long)blockIdx.x * blockDim.x + threadIdx.x;
    const int node = (int)(gt >> 5);
    const int lane = (int)(gt & 31);
    if (node >= Nn) return;
    const int c = lane * 4;

    const float4 v = *(const float4*)(g + (size_t)node * DCH + c);
    float p0 = v.x * W1[(c + 0) * 2 + 0] + v.y * W1[(c + 1) * 2 + 0] +
               v.z * W1[(c + 2) * 2 + 0] + v.w * W1[(c + 3) * 2 + 0];
    float p1 = v.x * W1[(c + 0) * 2 + 1] + v.y * W1[(c + 1) * 2 + 1] +
               v.z * W1[(c + 2) * 2 + 1] + v.w * W1[(c + 3) * 2 + 1];
    #pragma unroll
    for (int off = 16; off > 0; off >>= 1) {
        p0 += __shfl_xor(p0, off, 32);
        p1 += __shfl_xor(p1, off, 32);
    }
    if (lane == 0) {
        out[(size_t)node * 2 + 0] = p0 + b1[0];
        out[(size_t)node * 2 + 1] = p1 + b1[1];
    }
}

extern "C" void kernel_launch(void* const* d_in, const int* in_sizes, int n_in,
                              void* d_out, int out_size, void* d_ws, size_t ws_size,
                              hipStream_t stream) {
    const float* x        = (const float*)d_in[0];
    const int*   ei_phys  = (const int*)  d_in[1];
    const int*   ei_reg   = (const int*)  d_in[2];
    const float* W_conv   = (const float*)d_in[3];  // [L,2,128,128]
    const float* b_conv   = (const float*)d_in[4];  // [L,2,128]
    const float* ln_gamma = (const float*)d_in[5];  // [L,128]
    const float* ln_beta  = (const float*)d_in[6];  // [L,128]
    const float* head_W0  = (const float*)d_in[7];  // [128,128]
    const float* head_b0  = (const float*)d_in[8];  // [128]
    const float* head_W1  = (const float*)d_in[9];  // [128,2]
    const float* head_b1  = (const float*)d_in[10]; // [2]

    const int Nn = in_sizes[0] / DCH;      // 100000
    const int E  = in_sizes[1] / 2;        // 800000
    const size_t ND = (size_t)Nn * DCH;

    float* bufA = (float*)d_ws;            // h (layer output)
    float* bufB = bufA + ND;               // hw / head intermediate
    float* bufC = bufB + ND;               // scatter accumulator

    const int mtiles = (Nn + 15) / 16;
    const dim3 gemmGrid((mtiles + 7) / 8), blk256(256);
    const int scatterBlocks = (int)(((size_t)E * 32 + 255) / 256);
    const int nodeBlocks    = (Nn + 7) / 8;   // 32 threads per node

    const float* h_in = x;
    for (int l = 0; l < 2; ++l) {
        hipMemsetAsync(bufC, 0, ND * sizeof(float), stream);
        for (int t = 0; t < 2; ++t) {
            const float* Wl = W_conv + (size_t)(l * 2 + t) * DCH * DCH;
            gemm_f32_wmma<<<gemmGrid, blk256, 0, stream>>>(
                h_in, Wl, bufB, nullptr, 0, Nn);
            const int* ei = (t == 0) ? ei_phys : ei_reg;
            edge_scatter<<<scatterBlocks, blk256, 0, stream>>>(
                bufB, ei, ei + E, bufC, E);
        }
        ln_gelu_kernel<<<nodeBlocks, blk256, 0, stream>>>(
            bufC,
            b_conv + (size_t)(l * 2 + 0) * DCH,
            b_conv + (size_t)(l * 2 + 1) * DCH,
            ln_gamma + (size_t)l * DCH,
            ln_beta  + (size_t)l * DCH,
            bufA, Nn);
        h_in = bufA;
    }

    // head: gelu(h @ W0 + b0) fused into the WMMA GEMM epilogue
    gemm_f32_wmma<<<gemmGrid, blk256, 0, stream>>>(
        bufA, head_W0, bufB, head_b0, 1, Nn);
    head_out_kernel<<<nodeBlocks, blk256, 0, stream>>>(
        bufB, head_W1, head_b1, (float*)d_out, Nn);
}